// PredictionHead3D_37546604101870
// MI455X (gfx1250) — compile-verified
//
#include <hip/hip_runtime.h>
#include <hip/hip_bf16.h>

typedef float v2f __attribute__((ext_vector_type(2)));
typedef float v4f __attribute__((ext_vector_type(4)));
typedef float v8f __attribute__((ext_vector_type(8)));

#define HH 128
#define WW 128
#define NPIX (HH * WW)          // 16384 points = pixels
#define NBATCH 2
#define SEG_TH 0.7f
#define GAUSS_TH 0.7f
#define GEPS 1e-7f
#define PI_APPROX 3.14f
#define MASK_BIG 1e30f

// ---------------------------------------------------------------------------
// Kernel 1: build rank-8 (rank-6 + padding) point-feature table, K-permuted
// for the 16x4 f32 A-fragment lane layout:
//   slots [0..3] = K0,K1,K4,K5  (consumed by lanes 0-15)
//   slots [4..7] = K2,K3,K6,K7  (consumed by lanes 16-31)
// K mapping (paired with pixel features {i^2, i*j, j^2, i, j, 1, 0, 0}):
//   K0=a, K1=2b, K2=c, K3=lin_i, K4=lin_j, K5=const(+mask bias), K6=K7=0
// ---------------------------------------------------------------------------
__global__ void pg_point_features(const float* __restrict__ vm,
                                  const float* __restrict__ sm,
                                  float* __restrict__ pf) {
    int idx = blockIdx.x * blockDim.x + threadIdx.x;
    if (idx >= NBATCH * NPIX) return;
    int b = idx / NPIX;
    int p = idx - b * NPIX;
    float xc = (float)(p / WW);   // center row i
    float yc = (float)(p - (p / WW) * WW);  // center col j

    const float* v = vm + (size_t)b * 3 * NPIX;
    float v0  = v[p];
    float v1  = v[NPIX + p];
    float v2  = v[2 * NPIX + p];
    float seg = sm[(size_t)b * NPIX + p];

    float rh = fmaxf(v0, 0.0f) + 1.0f;
    float rw = fmaxf(v1, 0.0f) + 1.0f;
    float ih2 = 1.0f / (rh * rh);      // 1 / vh^2
    float iw2 = 1.0f / (rw * rw);      // 1 / vw^2
    float theta = PI_APPROX * (1.0f / (1.0f + __expf(-v2)));
    float s = __sinf(theta);
    float c = __cosf(theta);

    float a  = 0.5f * (c * c * ih2 + s * s * iw2);
    float bq = 0.5f * s * c * (iw2 - ih2);          // b
    float cc = 0.5f * (s * s * ih2 + c * c * iw2);

    float lin_i = -2.0f * a  * xc - 2.0f * bq * yc;
    float lin_j = -2.0f * bq * xc - 2.0f * cc * yc;
    float cst   = a * xc * xc + 2.0f * bq * xc * yc + cc * yc * yc;
    // masked points: push quad to +huge so exp(-quad) underflows to 0
    cst += (seg > SEG_TH) ? 0.0f : MASK_BIG;

    v4f lo = {a, 2.0f * bq, lin_j, cst};   // K0,K1,K4,K5
    v4f hi = {cc, lin_i, 0.0f, 0.0f};      // K2,K3,K6,K7
    v4f* out = (v4f*)(pf + (size_t)idx * 8);
    out[0] = lo;
    out[1] = hi;
}

// ---------------------------------------------------------------------------
// Kernel 2: Quad = PointFeat[16384x8] x PixelFeat[8x16384] via
// V_WMMA_F32_16X16X4_F32 (two chained K=4 steps), min-reduced over points.
// Block = 256 threads = 8 waves; each wave owns one 16-pixel N-tile and
// iterates over all 1024 M-tiles of points.
// ---------------------------------------------------------------------------
__global__ void __launch_bounds__(256)
pg_gauss_min_wmma(const float* __restrict__ pf, float* __restrict__ out) {
    const int wave  = threadIdx.x >> 5;
    const int lane  = threadIdx.x & 31;
    const int col   = lane & 15;          // N within tile
    const int khalf = lane >> 4;          // 0: K{0,1}/{4,5}, 1: K{2,3}/{6,7}

    const int blocksPerImage = NPIX / (8 * 16);          // 128
    const int b       = blockIdx.x / blocksPerImage;
    const int pixBase = (blockIdx.x - b * blocksPerImage) * 128 + wave * 16;
    const int N  = pixBase + col;
    const float fi = (float)(N / WW);
    const float fj = (float)(N - (N / WW) * WW);

    // B-fragment (pixel features), built in registers. Lane layout assumed to
    // mirror A: VGPR v of lane l holds K = v + 2*(l>=16), N = l%16.
    v2f b0, b1;
    b0.x = khalf ? (fj * fj) : (fi * fi);   // K2 : K0
    b0.y = khalf ? fi        : (fi * fj);   // K3 : K1
    b1.x = khalf ? 0.0f      : fj;          // K6 : K4
    b1.y = khalf ? 0.0f      : 1.0f;        // K7 : K5

    const v4f* apf = (const v4f*)(pf + (size_t)b * NPIX * 8);

    v8f acc = {3.0e38f, 3.0e38f, 3.0e38f, 3.0e38f,
               3.0e38f, 3.0e38f, 3.0e38f, 3.0e38f};

    const int ntiles = NPIX / 16;           // 1024 M-tiles
    #pragma unroll 4
    for (int t = 0; t < ntiles; ++t) {
        // lane fetches its 4 A-floats (both K-halves of its stripe) in one b128
        const v4f* aptr = apf + (size_t)(t * 16 + col) * 2 + khalf;
        v4f av = *aptr;
        __builtin_prefetch((const void*)(aptr + 16 * 2), 0, 0); // 1 tile ahead

        v2f a0 = {av.x, av.y};
        v2f a1 = {av.z, av.w};

        v8f d = {0.0f, 0.0f, 0.0f, 0.0f, 0.0f, 0.0f, 0.0f, 0.0f};
        d = __builtin_amdgcn_wmma_f32_16x16x4_f32(
                false, a0, false, b0, (short)0, d, false, false);
        d = __builtin_amdgcn_wmma_f32_16x16x4_f32(
                false, a1, false, b1, (short)0, d, false, false);

        acc[0] = fminf(acc[0], d[0]);
        acc[1] = fminf(acc[1], d[1]);
        acc[2] = fminf(acc[2], d[2]);
        acc[3] = fminf(acc[3], d[3]);
        acc[4] = fminf(acc[4], d[4]);
        acc[5] = fminf(acc[5], d[5]);
        acc[6] = fminf(acc[6], d[6]);
        acc[7] = fminf(acc[7], d[7]);
    }

    // in-lane reduction over the 8 M-rows this lane holds
    float m = fminf(fminf(fminf(acc[0], acc[1]), fminf(acc[2], acc[3])),
                    fminf(fminf(acc[4], acc[5]), fminf(acc[6], acc[7])));
    // combine the two lane halves (M=0..7 vs M=8..15) — wave32
    m = fminf(m, __shfl_xor(m, 16, 32));

    if (khalf == 0) {
        float g = __expf(-m + GEPS);        // exp(-min quad + eps)
        g = (g >= GAUSS_TH) ? g : 0.0f;     // threshold (all-masked -> 0)
        out[(size_t)b * NPIX + N] = g;
    }
}

extern "C" void kernel_launch(void* const* d_in, const int* in_sizes, int n_in,
                              void* d_out, int out_size, void* d_ws, size_t ws_size,
                              hipStream_t stream) {
    (void)in_sizes; (void)n_in; (void)out_size; (void)ws_size;
    const float* vm = (const float*)d_in[0];   // (2,3,128,128) f32
    const float* sm = (const float*)d_in[1];   // (2,1,128,128) f32
    // d_in[2] = size_of_slice: chunking only affects scan order; max/min is
    // associative, so it does not change the result. Ignored.
    float* pf  = (float*)d_ws;                 // NBATCH*NPIX*8 floats = 1 MiB
    float* out = (float*)d_out;                // (2,1,128,128) f32

    pg_point_features<<<(NBATCH * NPIX + 255) / 256, 256, 0, stream>>>(vm, sm, pf);
    pg_gauss_min_wmma<<<NBATCH * (NPIX / 128), 256, 0, stream>>>(pf, out);
}